// FirstOrderCondRNN_58858231824784
// MI455X (gfx1250) — compile-verified
//
#include <hip/hip_runtime.h>
#include <hip/hip_bf16.h>

typedef __attribute__((ext_vector_type(2))) float v2f;
typedef __attribute__((ext_vector_type(8))) float v8f;

#define N_KC   200
#define N_MBON 20
#define N_FBN  60
#define N_DAN  20
#define N_REC  100
#define N_RECP 112      // N padded to 7 tiles of 16 (K needs no padding: 25*4 == 100)
#define GB     16       // batches per workgroup
#define NWROW  (N_MBON * N_KC)   // 4000 floats of plastic state per batch
#define TAU_W  5.0f
#define TAU_R  1.0f
#define W_MAXV 0.05f

// Zeroed recurrent weight read: Wm = W_recur with rows [0,20) cols [80,100) zeroed.
__device__ __forceinline__ float wm_at(const float* __restrict__ Wrec, int j, int i) {
    if (j >= N_REC || i >= N_REC) return 0.f;
    if (j < N_MBON && i >= N_REC - N_DAN) return 0.f;
    return Wrec[j * N_REC + i];
}

__device__ __forceinline__ float clipw(float x) {
    return x < 0.f ? 0.f : (x > W_MAXV ? W_MAXV : x);
}

__global__ __launch_bounds__(256)
void mb_rnn_persistent(const float* __restrict__ rkc,   // (B,200,T)
                       const float* __restrict__ rext,  // (B,2,T)
                       const float* __restrict__ timev, // (T)
                       const float* __restrict__ W0,    // (B,20,200)
                       const float* __restrict__ wt0,   // (B,20,200)
                       const float* __restrict__ Wrec,  // (100,100)
                       const float* __restrict__ Wext,  // (60,2)
                       const float* __restrict__ Wro,   // (1,20)
                       const float* __restrict__ biasg, // (100)
                       float* __restrict__ rall,        // (T,B,100)
                       float* __restrict__ Wg,          // (B,20,200) state+output
                       float* __restrict__ wtg,         // (B,20,200) state+output
                       float* __restrict__ rout,        // (T,B)
                       int B, int T)
{
    __shared__ __align__(16) float s_rk [GB][N_KC];
    __shared__ __align__(16) float s_rbk[GB][N_KC];
    __shared__ __align__(16) float s_r  [2][GB][N_RECP]; // double-buffered rates
    __shared__ float s_I   [GB][N_RECP];
    __shared__ float s_rbd [GB][24];
    __shared__ float s_rdan[GB][24];
    __shared__ float s_bias[N_RECP];
    __shared__ float s_wro [32];
    __shared__ float s_wext[N_FBN][2];

    const int tid  = threadIdx.x;
    const int lane = tid & 31;
    const int wave = tid >> 5;
    const int n    = lane & 15;   // M index for A / N index for D
    const int h    = lane >> 4;   // half-wave selector
    const int B0   = blockIdx.x * GB;

    const float dt = timev[1] - timev[0];
    const float aW = dt / TAU_W;
    const float aR = dt / TAU_R;
    const int   NT = T - 1;

    // block-local float4 views of the plastic state (L2-resident working set)
    float4*       Wg4  = (float4*)(Wg  + (size_t)B0 * NWROW);
    float4*       wtg4 = (float4*)(wtg + (size_t)B0 * NWROW);
    const float4* W04  = (const float4*)(W0  + (size_t)B0 * NWROW);
    const float4* wt04 = (const float4*)(wt0 + (size_t)B0 * NWROW);

    // ---- init: copy plastic state into output regions, b128 transactions
    for (int i = tid; i < GB * (NWROW / 4); i += 256) {
        Wg4[i]  = W04[i];
        wtg4[i] = wt04[i];
    }
    for (int i = tid; i < N_RECP; i += 256) s_bias[i] = (i < N_REC) ? biasg[i] : 0.f;
    if (tid < 32) s_wro[tid] = (tid < N_MBON) ? Wro[tid] : 0.f;
    for (int i = tid; i < N_FBN * 2; i += 256) (&s_wext[0][0])[i] = Wext[i];
    for (int i = tid; i < GB * N_RECP; i += 256) {
        int b = i / N_RECP, j = i % N_RECP;
        float v = (j >= N_MBON && j < N_REC) ? 0.1f : 0.f;
        s_r[0][b][j] = v;
        s_r[1][b][j] = 0.f;
        s_I[b][j]    = 0.f;    // DAN segment [80,100) stays 0 forever
        if (j < N_REC) rall[(size_t)(B0 + b) * N_REC + j] = v;   // r_all[0]
    }
    for (int i = tid; i < GB * N_KC; i += 256) {
        int b = i / N_KC, k = i % N_KC;
        s_rbk[b][k] = rkc[((size_t)(B0 + b) * N_KC + k) * T + 0];
    }
    for (int i = tid; i < GB * 24; i += 256) {
        int b = i / 24, d = i % 24;
        s_rbd[b][d]  = (d < N_DAN) ? 0.1f : 0.f;
        s_rdan[b][d] = 0.f;
    }
    if (tid < GB) rout[B0 + tid] = 0.f;      // readout0: MBON rates start at 0

    // ---- preload B-fragments of Wm^T for this wave's tile (25 chunks: K == 100)
    v2f bf[25];
    if (wave < 7) {
        const int j = 16 * wave + n;         // N index (target neuron)
#pragma unroll
        for (int cc = 0; cc < 25; ++cc) {
            int i0 = 4 * cc + 2 * h;         // K index (source neuron)
            v2f bv;
            bv.x = wm_at(Wrec, j, i0);
            bv.y = wm_at(Wrec, j, i0 + 1);
            bf[cc] = bv;
        }
    }
    __syncthreads();

    // ================= time loop (sequential scan) =================
    for (int t = 0; t < NT; ++t) {
        const int cur = t & 1, nxt = cur ^ 1;

        // phase 1: stage rk_t (gather over time-minor layout), FBN external input
        for (int i = tid; i < GB * N_KC; i += 256) {
            int b = i / N_KC, k = i % N_KC;
            s_rk[b][k] = rkc[((size_t)(B0 + b) * N_KC + k) * T + t];
        }
        for (int i = tid; i < GB * N_FBN; i += 256) {
            int b = i / N_FBN, f = i % N_FBN;
            float e0 = rext[((size_t)(B0 + b) * 2 + 0) * T + t];
            float e1 = rext[((size_t)(B0 + b) * 2 + 1) * T + t];
            s_I[b][N_MBON + f] = s_wext[f][0] * e0 + s_wext[f][1] * e1;
        }
        __syncthreads();

        // phase 2: per-batch plastic matvec I_kc = W · rk
        // row-per-thread, contiguous 800B rows via b128 loads, no shuffles
        for (int row = tid; row < GB * N_MBON; row += 256) {
            int b = row / N_MBON, m = row % N_MBON;
            const float4* w4 = (const float4*)(Wg + (size_t)(B0 + b) * NWROW + m * N_KC);
            const float4* r4 = (const float4*)(&s_rk[b][0]);
            float acc = 0.f;
#pragma unroll 5
            for (int q = 0; q < N_KC / 4; ++q) {
                float4 w = w4[q];
                float4 r = r4[q];
                acc += w.x * r.x + w.y * r.y + w.z * r.z + w.w * r.w;
            }
            s_I[b][m] = acc;
        }
        __syncthreads();

        // phase 3: Wr = r · Wm^T via 25 chained f32 WMMAs per output tile
        if (wave < 7) {
            v8f acc = {};
#pragma unroll
            for (int cc = 0; cc < 25; ++cc) {
                int i0 = 4 * cc + 2 * h;
                v2f a;
                a.x = s_r[cur][n][i0];       // A: M = n (batch), K = i0
                a.y = s_r[cur][n][i0 + 1];
                acc = __builtin_amdgcn_wmma_f32_16x16x4_f32(
                        false, a, false, bf[cc], (short)0, acc, false, false);
            }
            const int j = 16 * wave + n;     // D: N = n → target neuron j
            if (j < N_REC) {
                float bj = s_bias[j];
#pragma unroll
                for (int v = 0; v < 8; ++v) {
                    int b = v + 8 * h;       // D: M rows = v, v+8
                    float pre  = acc[v] + bj + s_I[b][j];
                    float rel  = pre > 0.f ? pre : 0.f;
                    float rold = s_r[cur][b][j];
                    float rn   = rold + (rel - rold) * aR;
                    s_r[nxt][b][j] = rn;
                    if (j >= N_REC - N_DAN) s_rdan[b][j - (N_REC - N_DAN)] = rn;
                }
            }
        }
        __syncthreads();

        // phase 4: coalesced r_all store, readout, low-pass traces
        for (int i = tid; i < GB * N_REC; i += 256) {
            int b = i / N_REC, j = i % N_REC;
            rall[((size_t)(t + 1) * B + (B0 + b)) * N_REC + j] = s_r[nxt][b][j];
        }
        if (tid < GB) {
            float s = 0.f;
            for (int m = 0; m < N_MBON; ++m) s += s_r[nxt][tid][m] * s_wro[m];
            rout[(size_t)(t + 1) * B + B0 + tid] = s;
        }
        for (int v = tid; v < GB * (N_KC / 4); v += 256) {   // rbk, vectorized
            int b = v / (N_KC / 4), kv = v % (N_KC / 4);
            float4* rb4 = (float4*)(&s_rbk[b][0]);
            const float4* rk4 = (const float4*)(&s_rk[b][0]);
            float4 rb = rb4[kv], rk = rk4[kv];
            rb.x += (rk.x - rb.x) * aW;
            rb.y += (rk.y - rb.y) * aW;
            rb.z += (rk.z - rb.z) * aW;
            rb.w += (rk.w - rb.w) * aW;
            rb4[kv] = rb;
        }
        for (int i = tid; i < GB * N_DAN; i += 256) {
            int b = i / N_DAN, d = i % N_DAN;
            s_rbd[b][d] += (s_rdan[b][d] - s_rbd[b][d]) * aW;
        }
        __syncthreads();

        // phase 5: plasticity — stream wt/W through L2 as b128, update in place
        for (int v = tid; v < GB * (NWROW / 4); v += 256) {
            int b   = v / (NWROW / 4);
            int rem = v % (NWROW / 4);
            int d   = rem / (N_KC / 4);
            int kv  = rem % (N_KC / 4);
            float4 rk = ((const float4*)(&s_rk [b][0]))[kv];
            float4 rb = ((const float4*)(&s_rbk[b][0]))[kv];
            float  bd = s_rbd[b][d];
            float  rd = s_rdan[b][d];
            float4 wt = wtg4[v];
            wt.x += (bd * rk.x - rd * rb.x) * dt;
            wt.y += (bd * rk.y - rd * rb.y) * dt;
            wt.z += (bd * rk.z - rd * rb.z) * dt;
            wt.w += (bd * rk.w - rd * rb.w) * dt;
            wtg4[v] = wt;
            float4 Wv = Wg4[v];
            Wv.x = clipw(Wv.x + (wt.x - Wv.x) * aW);
            Wv.y = clipw(Wv.y + (wt.y - Wv.y) * aW);
            Wv.z = clipw(Wv.z + (wt.z - Wv.z) * aW);
            Wv.w = clipw(Wv.w + (wt.w - Wv.w) * aW);
            Wg4[v] = Wv;
        }
        __syncthreads();   // protects s_rk/W for next step's phases 1-2
    }
}

extern "C" void kernel_launch(void* const* d_in, const int* in_sizes, int n_in,
                              void* d_out, int out_size, void* d_ws, size_t ws_size,
                              hipStream_t stream) {
    const float* rkc   = (const float*)d_in[0];
    const float* rext  = (const float*)d_in[1];
    const float* timev = (const float*)d_in[2];
    const float* W0    = (const float*)d_in[3];
    const float* wt0   = (const float*)d_in[4];
    const float* Wrec  = (const float*)d_in[5];
    const float* Wext  = (const float*)d_in[6];
    const float* Wro   = (const float*)d_in[7];
    const float* biasg = (const float*)d_in[8];

    const int T = in_sizes[2];                       // 61
    const int B = in_sizes[0] / (N_KC * T);          // 2048

    float* out  = (float*)d_out;
    float* rall = out;                               // (T,B,100)
    float* Wf   = rall + (size_t)T * B * N_REC;      // (B,20,200)
    float* wtf  = Wf   + (size_t)B * NWROW;          // (B,20,200)
    float* rout = wtf  + (size_t)B * NWROW;          // (T,B)

    dim3 grid(B / GB);
    mb_rnn_persistent<<<grid, 256, 0, stream>>>(
        rkc, rext, timev, W0, wt0, Wrec, Wext, Wro, biasg,
        rall, Wf, wtf, rout, B, T);
}